// SelfAttentionDtwBias_88725434401238
// MI455X (gfx1250) — compile-verified
//
#include <hip/hip_runtime.h>

#define Bb 8
#define Hh 8
#define Ss 1024
#define Dd 64
#define Cc 21

typedef float v2f __attribute__((ext_vector_type(2)));
typedef float v8f __attribute__((ext_vector_type(8)));

// ---------------------------------------------------------------------------
// Kernel 1: per-head DTW bias:  bias[b,h,q,k] = sum_c dtw[b,c,q,k]*W[h,c]+b[h]
// Written into the attention region of d_out (used as scratch, overwritten by
// kernel 2 with the final attention). Reads each dtw element exactly once.
// ---------------------------------------------------------------------------
__global__ __launch_bounds__(256) void dtw_bias_kernel(
    const float* __restrict__ dtw, const float* __restrict__ W,
    const float* __restrict__ bvec, float* __restrict__ att)
{
  __shared__ float sW[Hh * Cc];
  __shared__ float sb[Hh];
  int tid = threadIdx.x;
  if (tid < Hh * Cc) sW[tid] = W[tid];
  if (tid < Hh)      sb[tid] = bvec[tid];
  __syncthreads();

  int wg = blockIdx.x;            // B * 64 qtiles * 4 kchunks = 2048
  int kc = wg & 3;
  int qt = (wg >> 2) & 63;
  int bb = wg >> 8;
  int q0 = qt * 16, k0 = kc * 256;

  for (int e = tid; e < 16 * 256; e += 256) {
    int row = e >> 8;
    int col = e & 255;
    int q = q0 + row, k = k0 + col;
    float acc[Hh];
#pragma unroll
    for (int h = 0; h < Hh; ++h) acc[h] = sb[h];
    const float* dp = dtw + ((size_t)bb * Cc * Ss + q) * Ss + k;
#pragma unroll
    for (int c = 0; c < Cc; ++c) {
      float v = dp[(size_t)c * Ss * Ss];
#pragma unroll
      for (int h = 0; h < Hh; ++h) acc[h] += v * sW[h * Cc + c];
    }
#pragma unroll
    for (int h = 0; h < Hh; ++h)
      att[(((size_t)(bb * Hh + h) * Ss + q) << 10) + k] = acc[h];
  }
}

// ---------------------------------------------------------------------------
// Kernel 2: per (b,h,qtile) fused scores(WMMA) + bias + mask + softmax +
// attention store + output = P@V (WMMA).  4 waves / 128 threads per WG.
// ---------------------------------------------------------------------------
__global__ __launch_bounds__(128) void attn_kernel(
    const float* __restrict__ Q, const float* __restrict__ K,
    const float* __restrict__ V, const int* __restrict__ mask,
    float* __restrict__ out, float* __restrict__ att)
{
  // 16 rows x 1024 cols score strip; pad 8 floats so row stride = 1032
  // (1032 mod 64 = 8 -> consecutive rows land on disjoint bank groups).
  __shared__ float sP[16][1032];

  int wg = blockIdx.x;            // (b*H + h)*64 + qtile = 4096
  int qt = wg & 63;
  int bh = wg >> 6;               // b*H + h
  int bb = bh >> 3;
  int q0 = qt * 16;

  int tid  = threadIdx.x;
  int lane = tid & 31;
  int wave = tid >> 5;
  int Ml   = lane & 15;           // M (or N) index within tile
  int kh   = lane >> 4;           // K-half select: lanes 16-31 hold K+2

  // ---- preload Q tile as WMMA A-fragments: A[M][4kk + 2*kh + {0,1}] ----
  const float* Qb = Q + ((size_t)bh * Ss + q0 + Ml) * Dd + kh * 2;
  v2f aq[16];
#pragma unroll
  for (int kk = 0; kk < 16; ++kk)
    aq[kk] = *(const v2f*)(Qb + 4 * kk);

  const size_t attRow = ((size_t)bh * Ss + q0) << 10;  // base of att[b,h,q0,0]

  // ---- Phase A: scores = QK^T/8 + bias, masked, into LDS ----
  for (int t = wave; t < 64; t += 4) {
    int n0 = t * 16;
    const float* Kb = K + ((size_t)bh * Ss + n0 + Ml) * Dd + kh * 2;
    v8f c = {};
#pragma unroll
    for (int kk = 0; kk < 16; ++kk) {
      v2f bk = *(const v2f*)(Kb + 4 * kk);
      c = __builtin_amdgcn_wmma_f32_16x16x4_f32(false, aq[kk], false, bk,
                                                (short)0, c, false, false);
    }
#pragma unroll
    for (int r = 0; r < 8; ++r) {
      int lq = r + 8 * kh;        // local q row of this C element
      size_t gi = attRow + ((size_t)lq << 10) + n0 + Ml;
      float s = c[r] * 0.125f + att[gi];
      int m = mask[(((size_t)(bb * Ss + q0 + lq)) << 10) + n0 + Ml];
      sP[lq][n0 + Ml] = (m == 0) ? -1e9f : s;
    }
  }
  __syncthreads();

  // ---- Phase B: row softmax (8 lanes per row, shuffle reductions) ----
  {
    int row = tid >> 3;
    int sub = tid & 7;
    float mx = -3.4e38f;
    for (int j = sub; j < Ss; j += 8) mx = fmaxf(mx, sP[row][j]);
    mx = fmaxf(mx, __shfl_xor(mx, 1));
    mx = fmaxf(mx, __shfl_xor(mx, 2));
    mx = fmaxf(mx, __shfl_xor(mx, 4));
    float sum = 0.f;
    for (int j = sub; j < Ss; j += 8) {
      float e = __expf(sP[row][j] - mx);
      sP[row][j] = e;
      sum += e;
    }
    sum += __shfl_xor(sum, 1);
    sum += __shfl_xor(sum, 2);
    sum += __shfl_xor(sum, 4);
    float inv = 1.0f / sum;
    for (int j = sub; j < Ss; j += 8) sP[row][j] *= inv;
  }
  __syncthreads();

  // ---- write attention, fully coalesced 128-wide sweeps ----
  for (int i = tid; i < 16 * 1024; i += 128) {
    int row = i >> 10, col = i & 1023;
    att[attRow + ((size_t)row << 10) + col] = sP[row][col];
  }

  // ---- Phase C: output tile [16 x 16] per wave = P[16,1024] @ V[1024,16] ----
  {
    int d0 = wave * 16;
    const float* Vb = V + (size_t)bh * Ss * Dd + d0 + Ml;
    v8f o = {};
#pragma unroll 4
    for (int kk = 0; kk < 256; ++kk) {
      int kb = 4 * kk + 2 * kh;
      v2f a = *(const v2f*)&sP[Ml][kb];          // A[M][K] from LDS
      v2f bv;
      bv.x = Vb[(size_t)kb * Dd];                // B[K][N] = V[k][d0+N]
      bv.y = Vb[(size_t)(kb + 1) * Dd];
      o = __builtin_amdgcn_wmma_f32_16x16x4_f32(false, a, false, bv,
                                                (short)0, o, false, false);
    }
#pragma unroll
    for (int r = 0; r < 8; ++r) {
      int lq = r + 8 * kh;
      out[((size_t)bh * Ss + q0 + lq) * Dd + d0 + Ml] = o[r];
    }
  }
}

// ---------------------------------------------------------------------------
extern "C" void kernel_launch(void* const* d_in, const int* in_sizes, int n_in,
                              void* d_out, int out_size, void* d_ws, size_t ws_size,
                              hipStream_t stream) {
  const float* Q    = (const float*)d_in[0];
  const float* K    = (const float*)d_in[1];
  const float* V    = (const float*)d_in[2];
  const float* dtw  = (const float*)d_in[3];
  const int*   mask = (const int*)d_in[4];
  const float* W    = (const float*)d_in[5];
  const float* bvec = (const float*)d_in[6];

  float* out = (float*)d_out;                       // [B,H,S,D]
  float* att = out + (size_t)Bb * Hh * Ss * Dd;     // [B,H,S,S]

  dtw_bias_kernel<<<Bb * 64 * 4, 256, 0, stream>>>(dtw, W, bvec, att);
  attn_kernel<<<Bb * Hh * 64, 128, 0, stream>>>(Q, K, V, mask, out, att);
}